// ReferenceMoELoRA_28587302322949
// MI455X (gfx1250) — compile-verified
//
#include <hip/hip_runtime.h>

// MoE-LoRA for MI455X (gfx1250, wave32). All GEMMs via V_WMMA_F32_16X16X4_F32
// (native fp32 matrix op -> reference precision, matrix pipe).
//
// Pipeline:
//   K1 router : g[t,k]  = dense gate matrix (0 except top-2, * softmax * alpha)
//   K2 h-GEMM : hv[t,n] = g[t, n/16] * sum_d x[t,d] * A[n/16, n%16, d]   (n = k*16+r)
//   K3 o-GEMM : out[t,d] = sum_n hv[t,n] * Bmat[n/16, d, n%16]
//
// K2/K3: 2Mx2N register tiles per wave (4 loads -> 4 WMMAs, 4 independent
// accumulation chains). Outer K-loop bound is a RUNTIME kernel argument so the
// compiler emits a compact loop (no full unroll / I$ blowup); the inner block
// of 4 K-steps is fully unrolled with immediate load offsets.

typedef __attribute__((ext_vector_type(2))) float v2f;
typedef __attribute__((ext_vector_type(8))) float v8f;

#define D_DIM  2048
#define K_EXP  16
#define R_RANK 16
#define NTOK   4096        // B*T
#define N_HV   256         // K_EXP * R_RANK

static __device__ __forceinline__ v8f wmma_f32(v2f a, v2f b, v8f c) {
  // D = A(16x4 f32) x B(4x16 f32) + C(16x16 f32)
  return __builtin_amdgcn_wmma_f32_16x16x4_f32(
      /*neg_a=*/false, a, /*neg_b=*/false, b,
      /*c_mod=*/(short)0, c, /*reuse_a=*/false, /*reuse_b=*/false);
}

// ---------------- Kernel 1: router + top2 + softmax -> dense gates -----------
__global__ __launch_bounds__(256) void moelora_router(
    const float* __restrict__ x, const float* __restrict__ Wr,
    const int* __restrict__ alpha_p, float* __restrict__ g)
{
  const int wave = threadIdx.x >> 5;
  const int lane = threadIdx.x & 31;
  const int t = blockIdx.x * 8 + wave;            // one wave per token

  float acc[K_EXP];
#pragma unroll
  for (int k = 0; k < K_EXP; ++k) acc[k] = 0.f;

  const float* xrow = x + (size_t)t * D_DIM;
  for (int d = lane; d < D_DIM; d += 32) {
    const float xv = xrow[d];
#pragma unroll
    for (int k = 0; k < K_EXP; ++k)
      acc[k] = fmaf(xv, Wr[k * D_DIM + d], acc[k]);
  }
#pragma unroll
  for (int k = 0; k < K_EXP; ++k) {
#pragma unroll
    for (int off = 16; off > 0; off >>= 1)
      acc[k] += __shfl_xor(acc[k], off, 32);
  }

  if (lane == 0) {
    const float alpha = (float)(*alpha_p);
    // argmax (ties -> lowest index, matches jax top_k)
    int i0 = 0; float v0 = acc[0];
#pragma unroll
    for (int k = 1; k < K_EXP; ++k) if (acc[k] > v0) { v0 = acc[k]; i0 = k; }
    int i1 = -1; float v1 = -3.4e38f;
#pragma unroll
    for (int k = 0; k < K_EXP; ++k)
      if (k != i0 && acc[k] > v1) { v1 = acc[k]; i1 = k; }
    // softmax over {v0, v1}, v0 >= v1
    const float e  = __expf(v1 - v0);
    const float dn = 1.f + e;
    const float g0 = alpha / dn;
    const float g1 = alpha * e / dn;
#pragma unroll
    for (int k = 0; k < K_EXP; ++k)
      g[t * K_EXP + k] = (k == i0) ? g0 : ((k == i1) ? g1 : 0.f);
  }
}

// ---------------- Kernel 2: hv = (x @ A^T) scaled by gates -------------------
// GEMM M=4096 (tokens), N=256 (k*16+r), Kd=2048.
// Block = 32 tokens (2 M-tiles) x all 16 N-tiles; each of 8 waves owns
// a 2Mx2N register tile (two experts). nchunk = D_DIM/16 (runtime arg).
__global__ __launch_bounds__(256) void moelora_hgemm(
    const float* __restrict__ x, const float* __restrict__ Amat,
    const float* __restrict__ g, float* __restrict__ hv, int nchunk)
{
  const int wave = threadIdx.x >> 5;
  const int lane = threadIdx.x & 31;
  const int m = lane & 15;        // row (A-op) / column (B-op) within tile
  const int h = lane >> 4;        // K-pair selector (lanes 16-31 carry K=2,3)
  const int row0 = blockIdx.x * 32;    // two 16-token M-tiles per block
  const int nt0  = wave * 2;           // this wave's two N-tiles (= experts)

  const float* xrow0 = x    + (size_t)(row0 + m)        * D_DIM + 2 * h;
  const float* xrow1 = x    + (size_t)(row0 + 16 + m)   * D_DIM + 2 * h;
  const float* arow0 = Amat + (size_t)(nt0 * 16 + m)       * D_DIM + 2 * h;
  const float* arow1 = Amat + (size_t)((nt0 + 1) * 16 + m) * D_DIM + 2 * h;

  v8f c00 = {}, c01 = {}, c10 = {}, c11 = {};

  for (int c = 0; c < nchunk; ++c) {         // runtime bound: compact loop
    const int base = c * 16;                 // 16 K-values per chunk
#pragma unroll
    for (int s = 0; s < 4; ++s) {            // 4 K-steps, immediate offsets
      const int kb = base + s * 4;
      const v2f a0 = *(const v2f*)(xrow0 + kb);
      const v2f a1 = *(const v2f*)(xrow1 + kb);
      const v2f b0 = *(const v2f*)(arow0 + kb);
      const v2f b1 = *(const v2f*)(arow1 + kb);
      c00 = wmma_f32(a0, b0, c00);
      c01 = wmma_f32(a0, b1, c01);
      c10 = wmma_f32(a1, b0, c10);
      c11 = wmma_f32(a1, b1, c11);
    }
  }

#pragma unroll
  for (int i = 0; i < 8; ++i) {
    const int ra = row0 + i + 8 * h;          // C/D: VGPR i holds M = i + 8*h
    const int rb = ra + 16;
    const float ga0 = g[ra * K_EXP + nt0];
    const float ga1 = g[ra * K_EXP + nt0 + 1];
    const float gb0 = g[rb * K_EXP + nt0];
    const float gb1 = g[rb * K_EXP + nt0 + 1];
    hv[(size_t)ra * N_HV + nt0 * 16 + m]       = c00[i] * ga0;
    hv[(size_t)ra * N_HV + (nt0 + 1) * 16 + m] = c01[i] * ga1;
    hv[(size_t)rb * N_HV + nt0 * 16 + m]       = c10[i] * gb0;
    hv[(size_t)rb * N_HV + (nt0 + 1) * 16 + m] = c11[i] * gb1;
  }
}

// ---------------- Kernel 3: out = hv @ Bbig ----------------------------------
// GEMM M=4096, N=2048 (d), Kd=256. Bbig[k*16+r, d] = Bmat[k, d, r]
// (r, r+1 contiguous in memory -> clean float2 fragment loads).
// Block = 32 tokens x 256 d-columns; each wave owns a 2Mx2N tile.
// Outer loop over experts (runtime bound nexp) -> inner body is pure
// pointer + immediate-offset loads (no per-step index math).
__global__ __launch_bounds__(256) void moelora_ogemm(
    const float* __restrict__ hv, const float* __restrict__ Bm,
    float* __restrict__ out, int nexp)
{
  const int wave = threadIdx.x >> 5;
  const int lane = threadIdx.x & 31;
  const int m = lane & 15;
  const int h = lane >> 4;
  const int row0 = blockIdx.x * 32;              // two M-tiles per block
  const int dt0  = blockIdx.y * 16 + wave * 2;   // two 16-col N-tiles per wave
  const int dcol0 = dt0 * 16 + m;
  const int dcol1 = (dt0 + 1) * 16 + m;

  const float* hr0 = hv + (size_t)(row0 + m)      * N_HV + 2 * h;
  const float* hr1 = hv + (size_t)(row0 + 16 + m) * N_HV + 2 * h;
  // B fragment base for expert 0: Bm[k=0, dcol, r=2h]
  const float* bb0 = Bm + (size_t)dcol0 * R_RANK + 2 * h;
  const float* bb1 = Bm + (size_t)dcol1 * R_RANK + 2 * h;

  v8f c00 = {}, c01 = {}, c10 = {}, c11 = {};

  for (int kk = 0; kk < nexp; ++kk) {        // runtime bound: compact loop
#pragma unroll
    for (int s = 0; s < 4; ++s) {            // r-steps: immediate offsets
      const v2f a0 = *(const v2f*)(hr0 + s * 4);
      const v2f a1 = *(const v2f*)(hr1 + s * 4);
      const v2f b0 = *(const v2f*)(bb0 + s * 4);
      const v2f b1 = *(const v2f*)(bb1 + s * 4);
      c00 = wmma_f32(a0, b0, c00);
      c01 = wmma_f32(a0, b1, c01);
      c10 = wmma_f32(a1, b0, c10);
      c11 = wmma_f32(a1, b1, c11);
    }
    hr0 += R_RANK;                 // next expert's 16 K-values in hv
    hr1 += R_RANK;
    bb0 += (size_t)D_DIM * R_RANK; // next expert's Bmat slab
    bb1 += (size_t)D_DIM * R_RANK;
  }

#pragma unroll
  for (int i = 0; i < 8; ++i) {
    const int ra = row0 + i + 8 * h;
    const int rb = ra + 16;
    out[(size_t)ra * D_DIM + dcol0] = c00[i];
    out[(size_t)ra * D_DIM + dcol1] = c01[i];
    out[(size_t)rb * D_DIM + dcol0] = c10[i];
    out[(size_t)rb * D_DIM + dcol1] = c11[i];
  }
}

extern "C" void kernel_launch(void* const* d_in, const int* in_sizes, int n_in,
                              void* d_out, int out_size, void* d_ws, size_t ws_size,
                              hipStream_t stream) {
  (void)in_sizes; (void)n_in; (void)out_size; (void)ws_size;
  const float* x     = (const float*)d_in[0];
  const int*   alpha = (const int*)  d_in[1];
  const float* A     = (const float*)d_in[2];
  const float* Bm    = (const float*)d_in[3];
  const float* Wr    = (const float*)d_in[4];
  float* out = (float*)d_out;

  float* g  = (float*)d_ws;               // [4096, 16]   = 256 KB
  float* hv = g + (size_t)NTOK * K_EXP;   // [4096, 256]  = 4 MB

  // K1: one wave per token, 8 waves/block -> 512 blocks
  moelora_router<<<NTOK / 8, 256, 0, stream>>>(x, Wr, alpha, g);
  // K2: one block per 32-token pair of M-tiles; 128 chunks of 16 K-values
  moelora_hgemm<<<NTOK / 32, 256, 0, stream>>>(x, A, g, hv, D_DIM / 16);
  // K3: 128 M-pairs x 8 column groups (256 cols each); 16 experts
  moelora_ogemm<<<dim3(NTOK / 32, D_DIM / 256), 256, 0, stream>>>(hv, Bm, out, K_EXP);
}